// WpLayer_79491254714439
// MI455X (gfx1250) — compile-verified
//
#include <hip/hip_runtime.h>
#include <stdint.h>

#define BATCH      8
#define SEQ_L      4096
#define FW         4
#define CH         512
#define XROWS      (SEQ_L + FW - 1)   /* 4099 */
#define ROWS       32                 /* output rows per block */
#define NATT       (ROWS + FW - 1)    /* 35 attention scalars per tile */
#define THREADS    128                /* 128 lanes x float4 = 512 channels */
#define ROWV4      (CH / 4)           /* row stride in float4 = 128 */

__global__ __launch_bounds__(THREADS)
void conv_attn_stream_kernel(const float* __restrict__ x,
                             const float* __restrict__ attn,
                             float* __restrict__ out)
{
    __shared__ float s_attn[NATT];

    const int t    = threadIdx.x;        // float4 column 0..127
    const int tile = blockIdx.x;         // 0..127
    const int b    = blockIdx.y;         // 0..7
    const int r0   = tile * ROWS;

    // ---- Stage attention window into LDS via gfx1250 async copy ----------
    if (t < NATT) {
        const float* gp = attn + (size_t)b * XROWS + (size_t)(r0 + t);
        uint64_t gaddr   = (uint64_t)(uintptr_t)gp;
        unsigned lds_off = (unsigned)(uintptr_t)(&s_attn[t]);
        asm volatile("global_load_async_to_lds_b32 %0, %1, off"
                     :: "v"(lds_off), "v"(gaddr) : "memory");
    }
    asm volatile("s_wait_asynccnt 0" ::: "memory");
    __syncthreads();

    // ---- Streaming sliding-window accumulation ---------------------------
    const float4* __restrict__ xp =
        (const float4*)(x + ((size_t)b * XROWS + (size_t)r0) * CH) + t;
    float4* __restrict__ op =
        (float4*)(out + ((size_t)b * SEQ_L + (size_t)r0) * CH) + t;

    // Prime the 4-deep window of products p[j] = attn[r0+j] * x[r0+j].
    float a0 = s_attn[0], a1 = s_attn[1], a2 = s_attn[2];
    float4 v;
    float4 p0, p1, p2;
    v = xp[0 * ROWV4]; p0.x = v.x*a0; p0.y = v.y*a0; p0.z = v.z*a0; p0.w = v.w*a0;
    v = xp[1 * ROWV4]; p1.x = v.x*a1; p1.y = v.y*a1; p1.z = v.z*a1; p1.w = v.w*a1;
    v = xp[2 * ROWV4]; p2.x = v.x*a2; p2.y = v.y*a2; p2.z = v.z*a2; p2.w = v.w*a2;

    #pragma unroll 8
    for (int i = 0; i < ROWS; ++i) {
        // Prefetch the stream ~16 rows ahead (once per 8 rows, clamped in-bounds).
        if ((i & 7) == 0) {
            int pr = r0 + i + 3 + 16;
            if (pr > XROWS - 1) pr = XROWS - 1;
            __builtin_prefetch((const void*)((const float4*)
                (x + ((size_t)b * XROWS + (size_t)pr) * CH) + t), 0, 3);
        }

        const float a3 = s_attn[i + 3];
        const float4 xv = xp[(i + 3) * ROWV4];

        float4 p3;
        p3.x = xv.x * a3; p3.y = xv.y * a3; p3.z = xv.z * a3; p3.w = xv.w * a3;

        float4 o;
        o.x = (p0.x + p1.x) + (p2.x + p3.x);
        o.y = (p0.y + p1.y) + (p2.y + p3.y);
        o.z = (p0.z + p1.z) + (p2.z + p3.z);
        o.w = (p0.w + p1.w) + (p2.w + p3.w);

        op[i * ROWV4] = o;

        p0 = p1; p1 = p2; p2 = p3;
    }
}

extern "C" void kernel_launch(void* const* d_in, const int* in_sizes, int n_in,
                              void* d_out, int out_size, void* d_ws, size_t ws_size,
                              hipStream_t stream)
{
    (void)in_sizes; (void)n_in; (void)d_ws; (void)ws_size; (void)out_size;
    const float* x    = (const float*)d_in[0];   // (8,1,4099,512) f32
    const float* attn = (const float*)d_in[1];   // (8,4099) f32
    float* out        = (float*)d_out;           // (8,1,4096,512) f32

    dim3 grid(SEQ_L / ROWS, BATCH);              // 128 x 8 = 1024 blocks
    conv_attn_stream_kernel<<<grid, THREADS, 0, stream>>>(x, attn, out);
}